// DistMultScore_12240656794088
// MI455X (gfx1250) — compile-verified
//
#include <hip/hip_runtime.h>
#include <stdint.h>

#define DIM 64
#define WAVES_PER_BLOCK 8
#define BLOCK_THREADS (WAVES_PER_BLOCK * 32)

// Fused DistMult edge kernel:
//   dr[e]  = sum_d head*rel*tail
//   h[dst] += sigmoid(dr[e]) * head*rel      (atomic segment-sum in L2)
// edge_emb is streamed via the CDNA5 async-to-LDS path (ASYNCcnt), double
// buffered; node_emb gathers and h atomics stay in the 192MB L2.
__global__ __launch_bounds__(BLOCK_THREADS) void distmult_fused_kernel(
    const float* __restrict__ node_emb,
    const float* __restrict__ edge_emb,
    const int* __restrict__ src,
    const int* __restrict__ dst,
    float* __restrict__ h,   // [N, 64], pre-zeroed
    float* __restrict__ dr,  // [E]
    int nEdges, int edgesPerSweep, int iters)
{
    // [buffer][wave-in-block][edge-of-pair][dim] : 8 KB
    __shared__ float smem[2][WAVES_PER_BLOCK][2][DIM];

    const int tid   = threadIdx.x;
    const int lane  = tid & 31;
    const int wid   = tid >> 5;
    const int sub   = lane >> 4;   // which edge of the wave's pair
    const int l16   = lane & 15;   // lane within 16-lane edge group
    const int gwave = blockIdx.x * WAVES_PER_BLOCK + wid;
    const int eoff  = gwave * 2 + sub;

    // Low 32 bits of a generic pointer to __shared__ == LDS byte address
    // (ISA 10.2).
    const uint32_t ldsA = (uint32_t)(uintptr_t)(&smem[0][wid][sub][l16 * 4]);
    const uint32_t ldsB = (uint32_t)(uintptr_t)(&smem[1][wid][sub][l16 * 4]);

    if (iters <= 0) return;

    // Prologue: stage iteration 0's 16B/lane slice of edge_emb into buffer A.
    {
        const int e0 = eoff;
        const int ec = (e0 < nEdges) ? e0 : (nEdges - 1);  // clamp: keep EXEC full
        const float* gp = edge_emb + (size_t)ec * DIM + l16 * 4;
        asm volatile("global_load_async_to_lds_b128 %0, %1, off"
                     :: "v"(ldsA), "v"((uint64_t)(uintptr_t)gp)
                     : "memory");
    }

    for (int it = 0; it < iters; ++it) {
        const uint32_t ldsNxt = (it & 1) ? ldsA : ldsB;
        const bool more = (it + 1) < iters;   // uniform across the grid

        // Drain prior ds_loads of the buffer we are about to overwrite before
        // the async engine can land new data there (WAR on LDS).
        asm volatile("s_wait_dscnt 0x0" ::: "memory");

        if (more) {
            const int en = (it + 1) * edgesPerSweep + eoff;
            const int ec = (en < nEdges) ? en : (nEdges - 1);
            const float* gp = edge_emb + (size_t)ec * DIM + l16 * 4;
            asm volatile("global_load_async_to_lds_b128 %0, %1, off"
                         :: "v"(ldsNxt), "v"((uint64_t)(uintptr_t)gp)
                         : "memory");
            asm volatile("s_wait_asynccnt 0x1" ::: "memory");  // oldest done
        } else {
            asm volatile("s_wait_asynccnt 0x0" ::: "memory");
        }

        const int e = it * edgesPerSweep + eoff;
        if (e < nEdges) {
            const float4 rel = *reinterpret_cast<const float4*>(
                &smem[it & 1][wid][sub][l16 * 4]);
            const int s = src[e];
            const int d = dst[e];
            const float4 hd = *reinterpret_cast<const float4*>(
                node_emb + (size_t)s * DIM + l16 * 4);
            const float4 tl = *reinterpret_cast<const float4*>(
                node_emb + (size_t)d * DIM + l16 * 4);

            float4 p;
            p.x = hd.x * rel.x;
            p.y = hd.y * rel.y;
            p.z = hd.z * rel.z;
            p.w = hd.w * rel.w;

            float part = p.x * tl.x + p.y * tl.y + p.z * tl.z + p.w * tl.w;
            // Reduce over the 16-lane edge group (masks 1/2/4/8 stay in-half
            // on wave32).
            part += __shfl_xor(part, 8, 32);
            part += __shfl_xor(part, 4, 32);
            part += __shfl_xor(part, 2, 32);
            part += __shfl_xor(part, 1, 32);

            if (l16 == 0) {
                // dr is write-once streamed output: NT store protects L2
                // residency of node_emb/h.
                __builtin_nontemporal_store(part, &dr[e]);
            }

            const float sig = 1.0f / (1.0f + expf(-part));
            float* out = h + (size_t)d * DIM + l16 * 4;
            atomicAdd(out + 0, sig * p.x);
            atomicAdd(out + 1, sig * p.y);
            atomicAdd(out + 2, sig * p.z);
            atomicAdd(out + 3, sig * p.w);
        }
    }
}

extern "C" void kernel_launch(void* const* d_in, const int* in_sizes, int n_in,
                              void* d_out, int out_size, void* d_ws, size_t ws_size,
                              hipStream_t stream) {
    (void)n_in; (void)out_size; (void)d_ws; (void)ws_size;

    const float* node_emb = (const float*)d_in[0];
    const float* edge_emb = (const float*)d_in[1];
    const int*   src      = (const int*)d_in[2];
    const int*   dst      = (const int*)d_in[3];

    const int E = in_sizes[2];          // number of edges
    const int N = in_sizes[0] / DIM;    // number of nodes

    float* h  = (float*)d_out;                       // [N, DIM]
    float* dr = (float*)d_out + (size_t)N * DIM;     // [E]

    // h accumulates via atomics; zero it inside the captured stream.
    hipMemsetAsync(h, 0, (size_t)N * DIM * sizeof(float), stream);

    const int blocks = 1024;                              // 8192 waves
    const int edgesPerSweep = blocks * WAVES_PER_BLOCK * 2;  // 16384 edges/sweep
    const int iters = (E + edgesPerSweep - 1) / edgesPerSweep;

    distmult_fused_kernel<<<blocks, BLOCK_THREADS, 0, stream>>>(
        node_emb, edge_emb, src, dst, h, dr, E, edgesPerSweep, iters);
}